// RoiPoolingConv_73821897884178
// MI455X (gfx1250) — compile-verified
//
#include <hip/hip_runtime.h>

#define POOL 14
#define NUM_ROIS 300
#define FH 100
#define FW 100
#define FC 512

typedef float v4f __attribute__((ext_vector_type(4)));
typedef int   v4i __attribute__((ext_vector_type(4)));

// TF2 half-pixel-center bilinear source coords, identical fp32 op order to the
// reference: src = (p+0.5)*(size/14) - 0.5 ; floor ; clamp ; +start.
__device__ __forceinline__ void axis_coords(int p, int start, int size, float scale,
                                            int& lo, int& hi, float& t) {
    float src = ((float)p + 0.5f) * scale - 0.5f;
    float f   = floorf(src);
    t         = src - f;
    int fi    = (int)f;
    int m     = size - 1;
    int c0 = fi     < 0 ? 0 : (fi     > m ? m : fi);
    int c1 = fi + 1 < 0 ? 0 : (fi + 1 > m ? m : fi + 1);
    lo = c0 + start;
    hi = c1 + start;
}

// One block per (roi, output row). 128 lanes x float4 = 512 channels.
// Double-buffered async global->LDS gathers (ASYNCcnt) pipeline the 4-corner
// fetch of pixel p+1 under the blend/store of pixel p. The x-axis coordinate
// table is computed once, vectorized across lanes, and extracted per pixel
// with v_readlane; column offsets ride in the 32-bit GVS vaddr so the per-
// pixel addressing is just two v_add_nc_u32.
__global__ __launch_bounds__(128)
void roi_align_row_kernel(const float* __restrict__ img,
                          const int*   __restrict__ rois,
                          float*       __restrict__ out) {
    __shared__ v4f tile[2][4][128];   // [buf][corner][lane] : 16 KB

    const int blk = blockIdx.x;       // = r*14 + py
    const int py  = blk % POOL;
    const int r   = blk / POOL;
    const int tid = threadIdx.x;

    const v4i roi = *(const v4i*)(rois + r * 4);   // {x, y, w, h}
    const float sx = (float)roi.z / (float)POOL;   // exact division, as reference
    const float sy = (float)roi.w / (float)POOL;

    // y-axis coords (scalar, once per block).
    int ylo, yhi; float ty;
    axis_coords(py, roi.y, roi.w, sy, ylo, yhi, ty);

    const char* rowlo = (const char*)img + (size_t)ylo * (FW * FC * 4);
    const char* rowhi = (const char*)img + (size_t)yhi * (FW * FC * 4);

    // x-axis table, one pixel per lane (lanes >= 14 compute unused junk; the
    // clamp keeps everything in range and nothing is dereferenced unless
    // read-laned with p < 14).
    int   xlo_v, xhi_v; float tx_v;
    axis_coords(tid, roi.x, roi.z, sx, xlo_v, xhi_v, tx_v);
    const int xlo_off_v = xlo_v * (FC * 4);        // byte column offsets
    const int xhi_off_v = xhi_v * (FC * 4);

    const unsigned voff = (unsigned)tid * 16u;     // per-lane channel offset

    // Loop-invariant LDS destinations (per-lane byte offsets).
    unsigned l[2][4];
#pragma unroll
    for (int b = 0; b < 2; ++b)
#pragma unroll
        for (int q = 0; q < 4; ++q)
            l[b][q] = (unsigned)(size_t)&tile[b][q][tid];

    const float onemty = 1.0f - ty;
    float* outrow = out + (size_t)blk * (POOL * FC);

    // Issue the 4 async corner gathers for pixel p into buffer `buf`.
    // GVS addressing: mem = saddr(row base, SGPR pair) + vaddr(lane offset).
    auto issue = [&](int p, int buf) {
        unsigned clo = (unsigned)__builtin_amdgcn_readlane(xlo_off_v, p);
        unsigned chi = (unsigned)__builtin_amdgcn_readlane(xhi_off_v, p);
        unsigned vlo = voff + clo;
        unsigned vhi = voff + chi;
        asm volatile("global_load_async_to_lds_b128 %0, %1, %2"
                     :: "v"(l[buf][0]), "v"(vlo), "s"(rowlo) : "memory");
        asm volatile("global_load_async_to_lds_b128 %0, %1, %2"
                     :: "v"(l[buf][1]), "v"(vhi), "s"(rowlo) : "memory");
        asm volatile("global_load_async_to_lds_b128 %0, %1, %2"
                     :: "v"(l[buf][2]), "v"(vlo), "s"(rowhi) : "memory");
        asm volatile("global_load_async_to_lds_b128 %0, %1, %2"
                     :: "v"(l[buf][3]), "v"(vhi), "s"(rowhi) : "memory");
    };

    issue(0, 0);

#pragma unroll
    for (int p = 0; p < POOL; ++p) {
        const int buf = p & 1;
        if (p + 1 < POOL) {
            issue(p + 1, buf ^ 1);
            // Async loads complete in order: <=4 outstanding => pixel p landed.
            asm volatile("s_wait_asynccnt 4" ::: "memory");
        } else {
            asm volatile("s_wait_asynccnt 0" ::: "memory");
        }
        // No __syncthreads: each lane reads back only the slots it wrote.

        const float tx = __builtin_bit_cast(
            float, __builtin_amdgcn_readlane(__builtin_bit_cast(int, tx_v), p));

        const float w00 = onemty * (1.0f - tx);
        const float w01 = onemty * tx;
        const float w10 = ty * (1.0f - tx);
        const float w11 = ty * tx;

        v4f a = tile[buf][0][tid];
        v4f b = tile[buf][1][tid];
        v4f c = tile[buf][2][tid];
        v4f d = tile[buf][3][tid];

        // Same left-to-right summation order as the reference.
        v4f o = w00 * a + w01 * b + w10 * c + w11 * d;

        v4f* op = (v4f*)(outrow + p * FC) + tid;
        __builtin_nontemporal_store(o, op);   // 120 MB streamed once
    }
}

extern "C" void kernel_launch(void* const* d_in, const int* in_sizes, int n_in,
                              void* d_out, int out_size, void* d_ws, size_t ws_size,
                              hipStream_t stream) {
    const float* img  = (const float*)d_in[0];   // (1,100,100,512) f32
    const int*   rois = (const int*)d_in[1];     // (1,300,4) i32
    float*       out  = (float*)d_out;           // (1,300,14,14,512) f32

    const int nblocks = NUM_ROIS * POOL;         // 4200 blocks, one output row each
    roi_align_row_kernel<<<dim3(nblocks), dim3(128), 0, stream>>>(img, rois, out);
}